// EnGMPHD_61117384622761
// MI455X (gfx1250) — compile-verified
//
#include <hip/hip_runtime.h>
#include <cmath>

typedef __attribute__((ext_vector_type(2))) float v2f;
typedef __attribute__((ext_vector_type(8))) float v8f;

#define LOG2PI_F 1.8378770664093453f
#define PD_CONST 0.98f
#define CLUTTER_CONST 1e-4f
#define LOCL 3.0

// ---------------------------------------------------------------------------
// Kernel A: mixture bandwidth covariance P = bw * (emp_cov .* loc)   [6x6]
// Single workgroup (8 waves). Phase 1: double-precision column means.
// Phase 2: WMMA f32 16x16x4 Gram of centered data (vars 0..5 live, rest 0).
// Deterministic: fixed wave->K-chunk mapping, fixed-order LDS combine.
// ---------------------------------------------------------------------------
__global__ void __launch_bounds__(256) engmphd_stats(
    const float* __restrict__ pm, float* __restrict__ Pout, int N, double bw) {
  __shared__ double sbuf[256];
  __shared__ float muLds[8];
  __shared__ float red[2048];
  const int tid = threadIdx.x;

  // ---- phase 1: column means (double accumulation, fixed-order tree) ----
  double acc[6] = {0, 0, 0, 0, 0, 0};
  for (int s = tid; s < N; s += 256) {
#pragma unroll
    for (int v = 0; v < 6; ++v) acc[v] += (double)pm[s * 6 + v];
  }
#pragma unroll
  for (int v = 0; v < 6; ++v) {
    sbuf[tid] = acc[v];
    __syncthreads();
    for (int off = 128; off > 0; off >>= 1) {
      if (tid < off) sbuf[tid] += sbuf[tid + off];
      __syncthreads();
    }
    if (tid == 0) muLds[v] = (float)(sbuf[0] / (double)N);
    __syncthreads();
  }

  // ---- phase 2: WMMA Gram over centered data ----
  const int lane = tid & 31;
  const int wave = tid >> 5;
  const int l = lane & 15;   // M (A) / N (B) index: variable 0..5 live
  const int h = lane >> 4;   // K half-select per documented layouts
  v8f cacc = {};
  const int nch = (N + 3) >> 2;  // chunks of K=4 samples
  for (int c = wave; c < nch; c += 8) {
    int s0 = c * 4 + 2 * h;
    float a0 = 0.0f, a1 = 0.0f;
    if (l < 6) {
      if (s0 < N)     a0 = pm[s0 * 6 + l] - muLds[l];
      if (s0 + 1 < N) a1 = pm[(s0 + 1) * 6 + l] - muLds[l];
    }
    v2f ab;
    ab.x = a0;
    ab.y = a1;
    // Gram: A = X_c^T (16x4), B = X_c (4x16) -> same register image.
    cacc = __builtin_amdgcn_wmma_f32_16x16x4_f32(
        false, ab, false, ab, (short)0, cacc, false, false);
  }
#pragma unroll
  for (int j = 0; j < 8; ++j) red[wave * 256 + j * 32 + lane] = cacc[j];
  __syncthreads();
  if (wave == 0) {
#pragma unroll
    for (int j = 0; j < 8; ++j) {
      float a = 0.0f;
      for (int w = 0; w < 8; ++w) a += red[w * 256 + j * 32 + lane];
      red[j * 32 + lane] = a;  // each (j,lane) slot owned by one thread
    }
  }
  __syncthreads();
  // C/D layout: VGPR j, lanes 0..15 -> (M=j, N=lane). We need M,N in 0..5.
  if (tid < 36) {
    int i = tid / 6, j = tid % 6;
    double emp = (double)red[i * 32 + j] / (double)(N - 1);
    double d = (double)(i - j);
    double loc = exp(-(d * d) / (2.0 * LOCL * LOCL));
    Pout[i * 6 + j] = (float)(bw * emp * loc);
  }
}

// ---------------------------------------------------------------------------
// Kernel B: per-component linearization + per-measurement points/log-terms.
// One thread per component n; loops over M measurements (staged in LDS).
// Writes: post_cov -> ws (once per n); points -> out_e; log_det_terms -> out_w.
// ---------------------------------------------------------------------------
__global__ void __launch_bounds__(256) engmphd_percomp(
    const float* __restrict__ pm, const float* __restrict__ pw,
    const float* __restrict__ meas, const float* __restrict__ Rm,
    const int* __restrict__ mask, const float* __restrict__ Pws,
    float* __restrict__ out_e, float* __restrict__ out_w,
    float* __restrict__ pc, int N, int M) {
  __shared__ float zsh[768];  // up to 256 measurements * 3
  __shared__ int msh[256];
  const int tid = threadIdx.x;
  for (int i = tid; i < M * 3; i += 256) zsh[i] = meas[i];
  for (int i = tid; i < M; i += 256) msh[i] = mask[i];
  __syncthreads();

  const int n = blockIdx.x * 256 + tid;
  if (n >= N) return;

  float P[36];
#pragma unroll
  for (int i = 0; i < 36; ++i) P[i] = Pws[i];
  float mean[6];
#pragma unroll
  for (int i = 0; i < 6; ++i) mean[i] = pm[n * 6 + i];

  // h(x) and analytic Jacobian (position block only; velocity cols are 0)
  const float x = mean[0], y = mean[1], z = mean[2];
  const float r2xy = x * x + y * y;
  const float rho2 = r2xy + z * z;
  const float rho = sqrtf(rho2);
  const float rxy = sqrtf(r2xy);
  float ybar0 = rho;
  float ybar1 = atan2f(y, x);
  float ybar2 = asinf(z / rho);
  float H[9];  // H[k][i], k meas-row, i pos-col
  H[0] = x / rho;        H[1] = y / rho;        H[2] = z / rho;
  H[3] = -y / r2xy;      H[4] = x / r2xy;       H[5] = 0.0f;
  const float ir2r = 1.0f / (rho2 * rxy);
  H[6] = -z * x * ir2r;  H[7] = -z * y * ir2r;  H[8] = rxy / rho2;

  // PHt[i][k] = sum_{j<3} P[i][j] * H[k][j]
  float PHt[18];
#pragma unroll
  for (int i = 0; i < 6; ++i)
#pragma unroll
    for (int k = 0; k < 3; ++k)
      PHt[i * 3 + k] =
          P[i * 6 + 0] * H[k * 3 + 0] + P[i * 6 + 1] * H[k * 3 + 1] +
          P[i * 6 + 2] * H[k * 3 + 2];

  // S = H * PHt + R (3x3)
  float S[9];
#pragma unroll
  for (int k = 0; k < 3; ++k)
#pragma unroll
    for (int lc = 0; lc < 3; ++lc)
      S[k * 3 + lc] = H[k * 3 + 0] * PHt[0 * 3 + lc] +
                      H[k * 3 + 1] * PHt[1 * 3 + lc] +
                      H[k * 3 + 2] * PHt[2 * 3 + lc] + Rm[k * 3 + lc];

  // symmetric 3x3 inverse + log|det|
  const float a = S[0], b = S[1], c = S[2], d = S[4], e = S[5], f = S[8];
  const float c00 = d * f - e * e;
  const float c01 = c * e - b * f;
  const float c02 = b * e - d * c;
  const float c11 = a * f - c * c;
  const float c12 = c * b - a * e;
  const float c22 = a * d - b * b;
  const float det = a * c00 + b * c01 + c * c02;
  const float logdet = logf(fabsf(det));
  const float idet = 1.0f / det;
  float Si[9];
  Si[0] = c00 * idet; Si[1] = c01 * idet; Si[2] = c02 * idet;
  Si[3] = c01 * idet; Si[4] = c11 * idet; Si[5] = c12 * idet;
  Si[6] = c02 * idet; Si[7] = c12 * idet; Si[8] = c22 * idet;

  // K = PHt * Sinv
  float K[18];
#pragma unroll
  for (int i = 0; i < 6; ++i)
#pragma unroll
    for (int k = 0; k < 3; ++k)
      K[i * 3 + k] = PHt[i * 3 + 0] * Si[0 * 3 + k] +
                     PHt[i * 3 + 1] * Si[1 * 3 + k] +
                     PHt[i * 3 + 2] * Si[2 * 3 + k];

  // post_cov = P - K * (H P) = P[i][j] - sum_k K[i][k]*PHt[j][k]  (P sym)
#pragma unroll
  for (int i = 0; i < 6; ++i)
#pragma unroll
    for (int j = 0; j < 6; ++j)
      pc[(size_t)n * 36 + i * 6 + j] =
          P[i * 6 + j] - (K[i * 3 + 0] * PHt[j * 3 + 0] +
                          K[i * 3 + 1] * PHt[j * 3 + 1] +
                          K[i * 3 + 2] * PHt[j * 3 + 2]);

  const float base =
      logf(PD_CONST) + logf(pw[n]) - 0.5f * (logdet + 3.0f * LOG2PI_F);

  for (int m = 0; m < M; ++m) {
    const float r0 = ybar0 - zsh[m * 3 + 0];
    const float r1 = ybar1 - zsh[m * 3 + 1];
    const float r2 = ybar2 - zsh[m * 3 + 2];
    const float maha = r0 * (Si[0] * r0 + Si[1] * r1 + Si[2] * r2) +
                       r1 * (Si[3] * r0 + Si[4] * r1 + Si[5] * r2) +
                       r2 * (Si[6] * r0 + Si[7] * r1 + Si[8] * r2);
    out_w[(size_t)m * N + n] = base - 0.5f * maha;  // log_det_terms
    const bool on = (msh[m] != 0);
    const size_t eb = ((size_t)m * N + n) * 6;
#pragma unroll
    for (int i = 0; i < 6; ++i) {
      float pv = mean[i] - (K[i * 3 + 0] * r0 + K[i * 3 + 1] * r1 +
                            K[i * 3 + 2] * r2);
      out_e[eb + i] = on ? pv : 0.0f;
    }
  }
}

// ---------------------------------------------------------------------------
// Kernel C: per-measurement logsumexp over N components (deterministic tree).
// logden[m] = log(CLUTTER + exp(logsumexp_n(terms[m,n])))
// ---------------------------------------------------------------------------
__global__ void __launch_bounds__(256) engmphd_lse(
    const float* __restrict__ out_w, float* __restrict__ logden, int N) {
  __shared__ float sb[256];
  const int m = blockIdx.x;
  const int tid = threadIdx.x;
  const float* row = out_w + (size_t)m * N;
  float mx = -INFINITY;
  for (int n = tid; n < N; n += 256) mx = fmaxf(mx, row[n]);
  sb[tid] = mx;
  __syncthreads();
  for (int off = 128; off > 0; off >>= 1) {
    if (tid < off) sb[tid] = fmaxf(sb[tid], sb[tid + off]);
    __syncthreads();
  }
  mx = sb[0];
  __syncthreads();
  float s = 0.0f;
  for (int n = tid; n < N; n += 256) s += expf(row[n] - mx);
  sb[tid] = s;
  __syncthreads();
  for (int off = 128; off > 0; off >>= 1) {
    if (tid < off) sb[tid] += sb[tid + off];
    __syncthreads();
  }
  if (tid == 0) logden[m] = logf(CLUTTER_CONST + expf(mx) * sb[0]);
}

// ---------------------------------------------------------------------------
// Kernel D: normalized weights in place (read-then-write same element).
// ---------------------------------------------------------------------------
__global__ void __launch_bounds__(256) engmphd_norm(
    float* __restrict__ out_w, const float* __restrict__ logden,
    const int* __restrict__ mask, int N, int M) {
  size_t i = (size_t)blockIdx.x * 256 + threadIdx.x;
  if (i >= (size_t)M * N) return;
  int m = (int)(i / (size_t)N);
  float t = out_w[i];
  out_w[i] = (mask[m] != 0) ? expf(t - logden[m]) : 0.0f;
}

// ---------------------------------------------------------------------------
// Kernel E: broadcast post_cov (2.88 MB, L2-resident) to M copies, masked.
// float4-vectorized; dominates runtime (~184 MB of stores).
// ---------------------------------------------------------------------------
__global__ void __launch_bounds__(256) engmphd_bcast(
    const float4* __restrict__ src, float4* __restrict__ dst,
    const int* __restrict__ mask, int N, int M) {
  size_t t = (size_t)blockIdx.x * 256 + threadIdx.x;
  const size_t per = (size_t)N * 9;  // float4s per measurement copy
  const size_t tot = per * (size_t)M;
  if (t >= tot) return;
  size_t m = t / per;
  size_t r = t - m * per;
  float4 v = src[r];
  if (mask[m] == 0) { v.x = 0.0f; v.y = 0.0f; v.z = 0.0f; v.w = 0.0f; }
  dst[t] = v;
}

extern "C" void kernel_launch(void* const* d_in, const int* in_sizes, int n_in,
                              void* d_out, int out_size, void* d_ws,
                              size_t ws_size, hipStream_t stream) {
  const float* pm = (const float*)d_in[0];    // prior_means   [N,6]
  const float* pw = (const float*)d_in[1];    // prior_weights [N]
  const float* meas = (const float*)d_in[2];  // measurements  [M,3]
  const float* Rm = (const float*)d_in[3];    // R             [3,3]
  const int* mask = (const int*)d_in[4];      // mask          [M]

  const int N = in_sizes[0] / 6;
  const int M = in_sizes[2] / 3;

  float* out_e = (float*)d_out;                  // [M,N,6]
  float* out_w = out_e + (size_t)M * N * 6;      // [M,N]
  float* out_c = out_w + (size_t)M * N;          // [M,N,6,6]

  float* ws = (float*)d_ws;
  float* Pws = ws;                               // 36 floats (pad to 64)
  float* pc = ws + 64;                           // N*36 floats
  float* logden = pc + (size_t)N * 36;           // M floats

  const double bw = pow(4.0 / ((double)N * 8.0), 2.0 / 10.0);

  engmphd_stats<<<1, 256, 0, stream>>>(pm, Pws, N, bw);

  engmphd_percomp<<<(N + 255) / 256, 256, 0, stream>>>(
      pm, pw, meas, Rm, mask, Pws, out_e, out_w, pc, N, M);

  engmphd_lse<<<M, 256, 0, stream>>>(out_w, logden, N);

  const size_t nw = (size_t)M * N;
  engmphd_norm<<<(unsigned)((nw + 255) / 256), 256, 0, stream>>>(
      out_w, logden, mask, N, M);

  const size_t nf4 = (size_t)M * N * 9;
  engmphd_bcast<<<(unsigned)((nf4 + 255) / 256), 256, 0, stream>>>(
      (const float4*)pc, (float4*)out_c, mask, N, M);
}